// Convolution_vect_2439541424143
// MI455X (gfx1250) — compile-verified
//
#include <hip/hip_runtime.h>

typedef __attribute__((ext_vector_type(2)))  float v2f;
typedef __attribute__((ext_vector_type(8)))  float v8f;
typedef __attribute__((ext_vector_type(16))) float v16f;

#define NC 64
#define NI 16
#define K2C 120
#define K1C 8
#define K0C 4
#define WPB 8   // waves per block

// ---------------------------------------------------------------------------
// Precompute: W2A[c][y][x][i] = sum_k U2[x,y,i,k]*w2[k,c]   (64*4096 floats)
//             C1x[c][x][y]    = sum_k U1[x,y,k]*w1[k,c]     (64*256, x-major)
//             C0g[c][x]       = sum_k U0[x,k]*w0[k,c]       (64*16)
// ---------------------------------------------------------------------------
__global__ __launch_bounds__(256) void precompute_kernel(
    const float* __restrict__ U2, const float* __restrict__ U1,
    const float* __restrict__ U0, const float* __restrict__ w2,
    const float* __restrict__ w1, const float* __restrict__ w0,
    float* __restrict__ W2A, float* __restrict__ C1x, float* __restrict__ C0g)
{
    int gid = blockIdx.x * 256 + threadIdx.x;          // 0..262143
    int i = gid & 15, x = (gid >> 4) & 15, y = (gid >> 8) & 15, c = gid >> 12;
    const float* u2row = U2 + ((x * 16 + y) * 16 + i) * K2C;
    float s = 0.f;
    for (int k = 0; k < K2C; ++k) s += u2row[k] * w2[k * NC + c];
    W2A[gid] = s;   // flat == ((c*16+y)*16+x)*16+i

    if (gid < NC * 256) {          // C1x: [c][x][y] loads as a WMMA A-operand
        int cc = gid >> 8, xx = (gid >> 4) & 15, yy = gid & 15;
        float s1 = 0.f;
        for (int k = 0; k < K1C; ++k) s1 += U1[(xx * 16 + yy) * K1C + k] * w1[k * NC + cc];
        C1x[gid] = s1;
    }
    if (gid < NC * 16) {
        int cc = gid >> 4, xx = gid & 15;
        float s0 = 0.f;
        for (int k = 0; k < K0C; ++k) s0 += U0[xx * K0C + k] * w0[k * NC + cc];
        C0g[gid] = s0;
    }
}

// ---------------------------------------------------------------------------
// Main kernel. Per wave = 16-atom tile; channels pipelined 3-deep with
// GLOBAL_LOAD_ASYNC_TO_LDS_B128 (ASYNCcnt) for the 16KB W2 slice.
// Per (wave,channel): 16 y-tiles x 4 wmma_f32_16x16x4 (corr=2), 4 more WMMAs
// for the corr=1 bias, VALU + shfl_xor(16) for corr=0.
// All per-lane feature data lives in vector SSA values (constant-indexed);
// the runtime half-select h only ever appears in load addresses -> no scratch.
// ---------------------------------------------------------------------------
__global__ __launch_bounds__(256) void contract_kernel(
    const float* __restrict__ nf, const float* __restrict__ W2A,
    const float* __restrict__ C1x, const float* __restrict__ C0g,
    float* __restrict__ out, int ntiles)
{
    __shared__ float sW2[3][4096];   // 3-deep pipeline, 16 KB per stage

    const int tid  = threadIdx.x;
    const int wave = tid >> 5;
    const int lane = tid & 31;
    const int h    = lane >> 4;      // wave half: selects K pair / M half
    const int m    = lane & 15;      // A: M row; B/D: column (atom)

    int tile = blockIdx.x * WPB + wave;
    if (tile >= ntiles) tile = ntiles - 1;   // wave-uniform clamp keeps EXEC all-ones

    // this wave's 2KB share of each 16KB W2 stage: 4 x (32 lanes x 16B)
    const uint32_t chunk   = (uint32_t)(wave * 2048 + lane * 16);       // bytes
    const uint32_t ldsBase = (uint32_t)(uintptr_t)(&sW2[0][0]);

    // INST_OFFSET is added to BOTH the LDS and the global address, so one
    // (vdst, vaddr, saddr) triple + 4 immediate offsets covers the share.
#define ISSUE_W2(bufidx, chan)                                                  \
    do {                                                                        \
        uint32_t _d = ldsBase + (uint32_t)(bufidx) * 16384u + chunk;            \
        const float* _s = W2A + (size_t)(chan) * 4096;                          \
        asm volatile("global_load_async_to_lds_b128 %0, %1, %2 offset:0"        \
                     :: "v"(_d), "v"(chunk), "s"(_s) : "memory");               \
        asm volatile("global_load_async_to_lds_b128 %0, %1, %2 offset:512"      \
                     :: "v"(_d), "v"(chunk), "s"(_s) : "memory");               \
        asm volatile("global_load_async_to_lds_b128 %0, %1, %2 offset:1024"     \
                     :: "v"(_d), "v"(chunk), "s"(_s) : "memory");               \
        asm volatile("global_load_async_to_lds_b128 %0, %1, %2 offset:1536"     \
                     :: "v"(_d), "v"(chunk), "s"(_s) : "memory");               \
    } while (0)

    ISSUE_W2(0, 0);
    ISSUE_W2(1, 1);

    for (int c = 0; c < NC; ++c) {
        // async batches complete in order: <=4 outstanding => batch c landed
        // (batch c+1 may still be in flight). No later batch exists for the
        // final two iterations, so drain fully there.
        if (c < NC - 2) {
            asm volatile("s_wait_asynccnt 4" ::: "memory");
        } else {
            asm volatile("s_wait_asynccnt 0" ::: "memory");
        }
        __syncthreads();   // batch c visible to all waves; all waves are done
                           // reading buffer (c-1)%3 == (c+2)%3
        if (c + 2 < NC) ISSUE_W2((c + 2) % 3, c + 2);

        const float* sw = sW2[c % 3];

        // lane's atom feature row (atom = m; both halves mirror the same atom)
        const float* frow = nf + ((size_t)(tile * 16 + m) * NC + c) * NI;
        v16f frv = *(const v16f*)frow;             // f[a,t], constant t only
        v8f  fx  = *(const v8f*)(frow + 8 * h);    // f[a, 8h+r] for corr=0
        v2f b[4];
        #pragma unroll
        for (int s = 0; s < 4; ++s)
            b[s] = *(const v2f*)(frow + 4 * s + 2 * h);   // B operand slices

        float acc[8];
        #pragma unroll
        for (int r = 0; r < 8; ++r) acc[r] = 0.f;

        // corr=2: D[x,a] = sum_i W2A[c][y=t][x][i] * f[a,i]; acc[x] += D * f[a,t]
        #pragma unroll
        for (int t = 0; t < 16; ++t) {
            v8f d = {};
            #pragma unroll
            for (int s = 0; s < 4; ++s) {
                v2f av = *(const v2f*)(sw + (t * 16 + m) * 16 + 4 * s + 2 * h);
                d = __builtin_amdgcn_wmma_f32_16x16x4_f32(
                        false, av, false, b[s], (short)0, d, false, false);
            }
            float ft = frv[t];
            #pragma unroll
            for (int r = 0; r < 8; ++r) acc[r] += d[r] * ft;
        }

        // corr=1 bias as one more 16x16x16 GEMM: G[x,a] = sum_y c1[c,x,y] f[a,y]
        {
            v8f g = {};
            const float* c1row = C1x + c * 256 + m * 16;
            #pragma unroll
            for (int s = 0; s < 4; ++s) {
                v2f av = *(const v2f*)(c1row + 4 * s + 2 * h);
                g = __builtin_amdgcn_wmma_f32_16x16x4_f32(
                        false, av, false, b[s], (short)0, g, false, false);
            }
            #pragma unroll
            for (int r = 0; r < 8; ++r) acc[r] += g[r];
        }

        // corr=0: out = sum_x (c0[c,x] + acc2[a,x]) * f[a,x]; halves via shfl
        v8f c0v = *(const v8f*)(C0g + c * 16 + 8 * h);
        float partial = 0.f;
        #pragma unroll
        for (int r = 0; r < 8; ++r)
            partial += (c0v[r] + acc[r]) * fx[r];
        float total = partial + __shfl_xor(partial, 16, 32);
        if (h == 0)
            out[(size_t)(tile * 16 + m) * NC + c] = total;
    }
#undef ISSUE_W2
}

// ---------------------------------------------------------------------------
extern "C" void kernel_launch(void* const* d_in, const int* in_sizes, int n_in,
                              void* d_out, int out_size, void* d_ws, size_t ws_size,
                              hipStream_t stream) {
    const float* nf = (const float*)d_in[0];
    const float* U2 = (const float*)d_in[1];
    const float* U1 = (const float*)d_in[2];
    const float* U0 = (const float*)d_in[3];
    const float* w2 = (const float*)d_in[4];
    const float* w1 = (const float*)d_in[5];
    const float* w0 = (const float*)d_in[6];
    float* out = (float*)d_out;

    float* W2A = (float*)d_ws;          // 64*4096 floats = 1 MB
    float* C1x = W2A + NC * 4096;       // 64*256
    float* C0g = C1x + NC * 256;        // 64*16

    precompute_kernel<<<NC * 4096 / 256, 256, 0, stream>>>(
        U2, U1, U0, w2, w1, w0, W2A, C1x, C0g);

    const int n_atoms = in_sizes[0] / (NC * NI);   // 20000
    const int ntiles  = (n_atoms + 15) / 16;       // 1250
    const int nblocks = (ntiles + WPB - 1) / WPB;  // 157
    contract_kernel<<<nblocks, 256, 0, stream>>>(nf, W2A, C1x, C0g, out, ntiles);
}